// Encoder_11991548690655
// MI455X (gfx1250) — compile-verified
//
#include <hip/hip_runtime.h>

#define NN      100000
#define NE      1200000
#define DIM     64
#define NLAYERS 3
#define NGRAPHS 512
#define BN_EPS  1e-5f
#define LDW     68   // padded LDS row stride (floats) to avoid 16-way bank conflicts

typedef __attribute__((ext_vector_type(2))) float v2f;
typedef __attribute__((ext_vector_type(8))) float v8f;

__device__ __forceinline__ void atomicAddF(float* p, float v) {
    __hip_atomic_fetch_add(p, v, __ATOMIC_RELAXED, __HIP_MEMORY_SCOPE_AGENT);
}

__global__ void zero_kernel(float* __restrict__ p, int n) {
    int i = blockIdx.x * 256 + threadIdx.x;
    if (i < n) p[i] = 0.0f;
}

// agg[n,d] = xin[n,d]   (xin may have row stride 64 or 192)
__global__ void init_agg_kernel(const float* __restrict__ xin, int xstride,
                                float* __restrict__ agg, int n) {
    int i = blockIdx.x * 256 + threadIdx.x;
    if (i < n) {
        int node = i >> 6, d = i & 63;
        agg[i] = xin[(size_t)node * xstride + d];
    }
}

// agg[dst,d] += x[src,d] * w[e]   — one thread per (edge, feature)
__global__ void scatter_kernel(const float* __restrict__ xin, int xstride,
                               const long long* __restrict__ ei,
                               const float* __restrict__ ew,
                               float* __restrict__ agg, int nE) {
    int i = blockIdx.x * 256 + threadIdx.x;
    if (i < nE * 64) {
        int e = i >> 6, d = i & 63;
        long long s = ei[e];        // row 0 of edge_index: src
        long long t = ei[nE + e];   // row 1 of edge_index: dst
        atomicAddF(&agg[t * 64 + d], xin[s * (long long)xstride + d] * ew[e]);
    }
}

// One workgroup = 16 nodes x full 64 dims. 4 waves, each owns a 16x16 output
// tile. h = relu(relu((agg)@W1^T + b1)@W2^T + b2), plus BN sum/sumsq partials.
// hout may alias agg: each block only reads/writes its own 16 rows, with all
// reads staged into LDS before any global store.
__global__ void __launch_bounds__(128)
gin_mlp_kernel(const float* agg,
               const float* __restrict__ W1, const float* __restrict__ b1,
               const float* __restrict__ W2, const float* __restrict__ b2,
               float* hout, float* __restrict__ stats) {
    __shared__ float lA[16 * LDW];
    __shared__ float lB[16 * LDW];

    const int tid  = threadIdx.x;       // 0..127
    const int lane = tid & 31;
    const int wave = tid >> 5;          // 0..3
    const int node0 = blockIdx.x * 16;

    // Stage 16x64 input tile into LDS (each thread: an eighth of a row = 8 floats)
    {
        const int row = tid >> 3;
        const int cb  = (tid & 7) * 8;
        const float* g = agg + (size_t)(node0 + row) * 64 + cb;
        float4 a = *(const float4*)(g);
        float4 b = *(const float4*)(g + 4);
        *(float4*)&lA[row * LDW + cb]     = a;
        *(float4*)&lB[row * LDW + cb]     = a; // scratch init (overwritten)
        *(float4*)&lA[row * LDW + cb + 4] = b;
    }
    __syncthreads();

    const int out0  = wave * 16;        // output-feature base of this wave
    const int mrow  = lane & 15;        // M row (A) == N col (B/C/D) for this lane
    const int khalf = (lane >> 4) << 1; // 0 or 2 (fp32 16x4 A striping)
    const int rbase = (lane >> 4) * 8;  // C/D row base

    // ---- GEMM1: mid = relu(A @ W1^T + b1) ----
    v8f c1 = {};
    {
        const float* w1row = W1 + (size_t)(out0 + mrow) * 64;
        #pragma unroll
        for (int kk = 0; kk < 16; ++kk) {
            const int kb = kk * 4 + khalf;
            v2f a; a.x = lA[mrow * LDW + kb]; a.y = lA[mrow * LDW + kb + 1];
            v2f b; b.x = w1row[kb];           b.y = w1row[kb + 1];
            c1 = __builtin_amdgcn_wmma_f32_16x16x4_f32(false, a, false, b,
                                                       (short)0, c1, false, false);
        }
    }
    __syncthreads();   // all lA/lB reads done before rewriting lB
    {
        const float bias = b1[out0 + mrow];
        #pragma unroll
        for (int g = 0; g < 8; ++g)
            lB[(rbase + g) * LDW + out0 + mrow] = fmaxf(c1[g] + bias, 0.0f);
    }
    __syncthreads();

    // ---- GEMM2: out = relu(mid @ W2^T + b2) ----
    v8f c2 = {};
    {
        const float* w2row = W2 + (size_t)(out0 + mrow) * 64;
        #pragma unroll
        for (int kk = 0; kk < 16; ++kk) {
            const int kb = kk * 4 + khalf;
            v2f a; a.x = lB[mrow * LDW + kb]; a.y = lB[mrow * LDW + kb + 1];
            v2f b; b.x = w2row[kb];           b.y = w2row[kb + 1];
            c2 = __builtin_amdgcn_wmma_f32_16x16x4_f32(false, a, false, b,
                                                       (short)0, c2, false, false);
        }
    }
    __syncthreads();   // lA reads (GEMM1) long done; safe to rewrite lA
    {
        const float bias = b2[out0 + mrow];
        #pragma unroll
        for (int g = 0; g < 8; ++g)
            lA[(rbase + g) * LDW + out0 + mrow] = fmaxf(c2[g] + bias, 0.0f);
    }
    __syncthreads();

    // Store h tile to global + BN partial sums
    {
        const int row = tid >> 3;
        const int cb  = (tid & 7) * 8;
        float* g = hout + (size_t)(node0 + row) * 64 + cb;
        *(float4*)(g)     = *(const float4*)&lA[row * LDW + cb];
        *(float4*)(g + 4) = *(const float4*)&lA[row * LDW + cb + 4];
    }
    if (tid < 64) {
        float s = 0.0f;
        #pragma unroll
        for (int r = 0; r < 16; ++r) s += lA[r * LDW + tid];
        atomicAddF(&stats[tid], s);
    } else {
        const int f = tid - 64;
        float s = 0.0f;
        #pragma unroll
        for (int r = 0; r < 16; ++r) { float v = lA[r * LDW + f]; s += v * v; }
        atomicAddF(&stats[64 + f], s);
    }
}

__global__ void bn_finalize_kernel(float* __restrict__ stats, float inv_n) {
    int d = threadIdx.x;  // 64 threads
    float mu  = stats[d] * inv_n;
    float var = stats[64 + d] * inv_n - mu * mu;
    stats[128 + d] = mu;
    stats[192 + d] = rsqrtf(var + BN_EPS);
}

// x = gamma*(h-mu)*rsig + beta ; write xs block ; atomic 'add' pooling
__global__ void bn_apply_pool_kernel(const float* __restrict__ h,
                                     const float* __restrict__ stats,
                                     const float* __restrict__ gamma,
                                     const float* __restrict__ beta,
                                     const long long* __restrict__ batch,
                                     float* __restrict__ out_xs,
                                     float* __restrict__ out_pool,
                                     int layer, int n) {
    int i = blockIdx.x * 256 + threadIdx.x;
    if (i < n) {
        int node = i >> 6, d = i & 63;
        float v = gamma[d] * (h[i] - stats[128 + d]) * stats[192 + d] + beta[d];
        size_t col = (size_t)layer * DIM + d;
        out_xs[(size_t)node * (NLAYERS * DIM) + col] = v;
        long long g = batch[node];
        atomicAddF(&out_pool[(size_t)g * (NLAYERS * DIM) + col], v);
    }
}

extern "C" void kernel_launch(void* const* d_in, const int* in_sizes, int n_in,
                              void* d_out, int out_size, void* d_ws, size_t ws_size,
                              hipStream_t stream) {
    const float*     x     = (const float*)d_in[0];
    const long long* ei    = (const long long*)d_in[1];   // int64 [2, E]
    const float*     ew    = (const float*)d_in[2];
    const long long* batch = (const long long*)d_in[3];   // int64 [N]
    // d_in[4] = pool (static 0 -> 'add' pooling)
    const float* W1    = (const float*)d_in[5];
    const float* b1    = (const float*)d_in[6];
    const float* W2    = (const float*)d_in[7];
    const float* b2    = (const float*)d_in[8];
    const float* gamma = (const float*)d_in[9];
    const float* beta  = (const float*)d_in[10];

    float* out_pool = (float*)d_out;                                   // [512, 192]
    float* out_xs   = out_pool + (size_t)NGRAPHS * NLAYERS * DIM;      // [N, 192]

    float* agg   = (float*)d_ws;               // N*64 floats (also holds h)
    float* stats = agg + (size_t)NN * DIM;     // 256 floats: sum|sumsq|mu|rsig

    const int feat = NN * DIM;                         // 6,400,000
    const int featBlocks = (feat + 255) / 256;
    const int poolElems = NGRAPHS * NLAYERS * DIM;     // 98,304

    zero_kernel<<<(poolElems + 255) / 256, 256, 0, stream>>>(out_pool, poolElems);

    for (int i = 0; i < NLAYERS; ++i) {
        const float* xin = (i == 0) ? x : (out_xs + (size_t)(i - 1) * DIM);
        const int xstride = (i == 0) ? DIM : NLAYERS * DIM;

        zero_kernel<<<1, 256, 0, stream>>>(stats, 256);
        init_agg_kernel<<<featBlocks, 256, 0, stream>>>(xin, xstride, agg, feat);
        scatter_kernel<<<(NE * DIM + 255) / 256, 256, 0, stream>>>(xin, xstride, ei, ew, agg, NE);
        gin_mlp_kernel<<<NN / 16, 128, 0, stream>>>(agg,
                                                    W1 + (size_t)i * DIM * DIM, b1 + i * DIM,
                                                    W2 + (size_t)i * DIM * DIM, b2 + i * DIM,
                                                    agg /* h aliases agg (row-local) */, stats);
        bn_finalize_kernel<<<1, 64, 0, stream>>>(stats, 1.0f / NN);
        bn_apply_pool_kernel<<<featBlocks, 256, 0, stream>>>(agg, stats,
                                                             gamma + i * DIM, beta + i * DIM,
                                                             batch, out_xs, out_pool, i, feat);
    }
}